// BeamSearch_13469017440344
// MI455X (gfx1250) — compile-verified
//
#include <hip/hip_runtime.h>
#include <hip/hip_bf16.h>
#include <stdint.h>

#define B_      8
#define L_      512
#define V_      32000
#define ROWS    (B_ * L_)
#define T5      5
#define K3      3
#define THREADS 320                  // 10 wave32 waves
#define CHUNK   (THREADS * 4)        // 1280 floats / chunk (5120 B)
#define NCHUNK  (V_ / CHUNK)         // 25, exact (no tail)
#define NBUF    4                    // async ring depth (4 chunks in flight)
#define PADN    512                  // padded reduction width
#define EPSF    2.220446049250313e-16f

// ---------------- CDNA5 async global->LDS path --------------------------
#if defined(__has_builtin)
#  if __has_builtin(__builtin_amdgcn_global_load_async_to_lds_b128)
#    define HAVE_ASYNC_BUILTIN 1
#  endif
#  if __has_builtin(__builtin_amdgcn_s_wait_asynccnt)
#    define HAVE_WAIT_ASYNC_BUILTIN 1
#  endif
#endif

// Builtin parameter types (from hipcc diagnostic): non-const pointer to a
// 16-byte int vector, qualified with the global / LDS address spaces.
typedef int v4i_ __attribute__((vector_size(16)));
typedef __attribute__((address_space(1))) v4i_* gas_v4i;   // global (AS1)
typedef __attribute__((address_space(3))) v4i_* las_v4i;   // LDS    (AS3)

__device__ __forceinline__ void async_copy16(const float* g, float* l) {
#if defined(HAVE_ASYNC_BUILTIN)
  __builtin_amdgcn_global_load_async_to_lds_b128(
      (gas_v4i)g, (las_v4i)l, /*imm offset*/0, /*cpol*/0);
#else
  // vdst = LDS byte offset VGPR; addr = 64-bit global address VGPR pair
  unsigned loff = (unsigned)(unsigned long long)(las_v4i)l;
  unsigned long long ga = (unsigned long long)g;
  asm volatile("global_load_async_to_lds_b128 %0, %1, off"
               :: "v"(loff), "v"(ga) : "memory");
#endif
}

#if defined(HAVE_WAIT_ASYNC_BUILTIN)
#  define WAIT_ASYNC(N) __builtin_amdgcn_s_wait_asynccnt(N)
#else
#  define WAIT_ASYNC(N) asm volatile("s_wait_asynccnt " #N ::: "memory")
#endif

// ---------------- Kernel 1: per-row online softmax + top-5 --------------
__global__ __launch_bounds__(THREADS)
void softmax_top5_kernel(const float* __restrict__ logits,
                         float* __restrict__ ws_lp,
                         int*   __restrict__ ws_ix) {
  __shared__ __align__(16) float tile[NBUF][CHUNK]; // async ring, 20 KB
  __shared__ float rm[PADN], rs[PADN];
  __shared__ float tv[PADN][T5];
  __shared__ int   ti[PADN][T5];

  const int tid = threadIdx.x;
  const int row = blockIdx.x;                      // row = b*L + l
  const float* rowp = logits + (size_t)row * V_;

  const float NEG_INF = -__builtin_inff();
  float m = NEG_INF, s = 0.0f;
  float v[T5]; int ix[T5];
#pragma unroll
  for (int j = 0; j < T5; ++j) { v[j] = NEG_INF; ix[j] = 0; }

  // prologue: fill pipeline with chunks 0..NBUF-2
#pragma unroll
  for (int c = 0; c < NBUF - 1; ++c)
    async_copy16(rowp + (size_t)c * CHUNK + tid * 4, &tile[c][tid * 4]);

  for (int c = 0; c < NCHUNK; ++c) {
    // keep NBUF chunks in flight; in-order completion => constant waits
    if (c + NBUF - 1 < NCHUNK) {
      async_copy16(rowp + (size_t)(c + NBUF - 1) * CHUNK + tid * 4,
                   &tile[(c + NBUF - 1) & (NBUF - 1)][tid * 4]);
      WAIT_ASYNC(3);                 // chunks c..c+3 outstanding -> c done
    } else if (c + 2 < NCHUNK) {
      WAIT_ASYNC(2);
    } else if (c + 1 < NCHUNK) {
      WAIT_ASYNC(1);
    } else {
      WAIT_ASYNC(0);
    }

    const float4 x4 =
        *reinterpret_cast<const float4*>(&tile[c & (NBUF - 1)][tid * 4]);
    const int base = c * CHUNK + tid * 4;
    float xs[4] = {x4.x, x4.y, x4.z, x4.w};
#pragma unroll
    for (int j = 0; j < 4; ++j) {
      float x = xs[j];
      // online logsumexp accumulator (1 transcendental per element)
      if (x > m) { s = s * __expf(m - x) + 1.0f; m = x; }
      else       { s += __expf(x - m); }
      // online sorted top-5 (descending); strict '>' keeps lower index on tie
      if (x > v[4]) {
        int gi = base + j;
        int p = 4;
#pragma unroll
        for (int q = 4; q > 0; --q) {
          if (x > v[q - 1]) { v[q] = v[q - 1]; ix[q] = ix[q - 1]; p = q - 1; }
        }
        v[p] = x; ix[p] = gi;
      }
    }
  }

  // ---- block reduction (padded to 512) ----
  rm[tid] = m; rs[tid] = s;
#pragma unroll
  for (int j = 0; j < T5; ++j) { tv[tid][j] = v[j]; ti[tid][j] = ix[j]; }
  if (tid < PADN - THREADS) {
    rm[THREADS + tid] = NEG_INF; rs[THREADS + tid] = 0.0f;
#pragma unroll
    for (int j = 0; j < T5; ++j) { tv[THREADS + tid][j] = NEG_INF; ti[THREADS + tid][j] = 0; }
  }
  __syncthreads();

  for (int off = PADN / 2; off >= 1; off >>= 1) {
    if (tid < off) {
      // merge (max, sum) with rescaling; slot tid is always a live lane
      float m1 = rm[tid], s1 = rs[tid];
      float m2 = rm[tid + off], s2 = rs[tid + off];
      float M  = fmaxf(m1, m2);
      float S  = s1 * __expf(m1 - M) + s2 * __expf(m2 - M);
      rm[tid] = M; rs[tid] = S;
      // merge two descending 5-lists -> top 5
      float av[T5], bv[T5]; int ai[T5], bi[T5];
#pragma unroll
      for (int j = 0; j < T5; ++j) {
        av[j] = tv[tid][j];       ai[j] = ti[tid][j];
        bv[j] = tv[tid + off][j]; bi[j] = ti[tid + off][j];
      }
      float ov[T5]; int oi[T5];
      int ia = 0, ib = 0;
#pragma unroll
      for (int k = 0; k < T5; ++k) {
        bool takeA = (ib >= T5) || (av[ia] >= bv[ib]);
        if (takeA) { ov[k] = av[ia]; oi[k] = ai[ia]; ++ia; }
        else       { ov[k] = bv[ib]; oi[k] = bi[ib]; ++ib; }
      }
#pragma unroll
      for (int j = 0; j < T5; ++j) { tv[tid][j] = ov[j]; ti[tid][j] = oi[j]; }
    }
    __syncthreads();
  }

  if (tid == 0) {
    float M = rm[0], S = rs[0];
#pragma unroll
    for (int j = 0; j < T5; ++j) {
      float p = __expf(tv[0][j] - M) / S;              // softmax prob of top-j
      ws_lp[(size_t)row * T5 + j] = __logf(p + EPSF);  // log(p + eps), as ref
      ws_ix[(size_t)row * T5 + j] = ti[0][j];
    }
  }
}

// ---------------- Kernel 2: per-batch beam scan + backtrack --------------
__global__ __launch_bounds__(32)
void beam_scan_kernel(const float* __restrict__ ws_lp,
                      const int*   __restrict__ ws_ix,
                      float* __restrict__ out) {
  __shared__ int tok[L_][K3];
  __shared__ int par[L_][K3];
  const int b = blockIdx.x;

  if (threadIdx.x == 0) {
    const float NEG_INF = -__builtin_inff();
    float sc[K3] = {0.0f, NEG_INF, NEG_INF};   // single initial beam

    for (int t = 0; t < L_; ++t) {
      const size_t r5 = ((size_t)b * L_ + t) * T5;
      float lp[T5]; int id[T5];
#pragma unroll
      for (int j = 0; j < T5; ++j) { lp[j] = ws_lp[r5 + j]; id[j] = ws_ix[r5 + j]; }

      // top-3 of 15 candidates, stable in flat index order (k*5+j)
      float bs[K3] = {NEG_INF, NEG_INF, NEG_INF};
      int   bk[K3] = {0, 0, 0}, bj[K3] = {0, 0, 0};
#pragma unroll
      for (int k = 0; k < K3; ++k) {
#pragma unroll
        for (int j = 0; j < T5; ++j) {
          float c = sc[k] + lp[j];
          if (c > bs[2]) {
            if (c > bs[0]) {
              bs[2] = bs[1]; bk[2] = bk[1]; bj[2] = bj[1];
              bs[1] = bs[0]; bk[1] = bk[0]; bj[1] = bj[0];
              bs[0] = c;     bk[0] = k;     bj[0] = j;
            } else if (c > bs[1]) {
              bs[2] = bs[1]; bk[2] = bk[1]; bj[2] = bj[1];
              bs[1] = c;     bk[1] = k;     bj[1] = j;
            } else {
              bs[2] = c;     bk[2] = k;     bj[2] = j;
            }
          }
        }
      }
#pragma unroll
      for (int r = 0; r < K3; ++r) {
        tok[t][r] = id[bj[r]];
        par[t][r] = bk[r];
        sc[r]     = bs[r];
      }
    }

    // backtrack parent pointers -> tokens[b, t, r]; then scores[b, r]
    for (int r = 0; r < K3; ++r) {
      int cur = r;
      for (int t = L_ - 1; t >= 0; --t) {
        out[((size_t)b * L_ + t) * K3 + r] = (float)tok[t][cur];
        cur = par[t][cur];
      }
      out[(size_t)B_ * L_ * K3 + (size_t)b * K3 + r] = sc[r];
    }
  }
}

// ---------------- host launcher -----------------------------------------
extern "C" void kernel_launch(void* const* d_in, const int* in_sizes, int n_in,
                              void* d_out, int out_size, void* d_ws, size_t ws_size,
                              hipStream_t stream) {
  (void)in_sizes; (void)n_in; (void)out_size; (void)ws_size;
  const float* logits = (const float*)d_in[0];
  float* ws_lp = (float*)d_ws;                                     // ROWS*5 f32
  int*   ws_ix = (int*)((char*)d_ws + sizeof(float) * ROWS * T5);  // ROWS*5 i32
  float* out   = (float*)d_out;

  softmax_top5_kernel<<<ROWS, THREADS, 0, stream>>>(logits, ws_lp, ws_ix);
  beam_scan_kernel<<<B_, 32, 0, stream>>>(ws_lp, ws_ix, out);
}